// MemoryOptimizedMomentHead_79413945303747
// MI455X (gfx1250) — compile-verified
//
#include <hip/hip_runtime.h>
#include <math.h>

// ---------------------------------------------------------------------------
// MemoryOptimizedMomentHead for MI455X (gfx1250, wave32, WMMA)
// fp32 matrix pipe (V_WMMA_F32_16X16X4_F32) for all matmuls; the memory-bound
// W2 head GEMM uses double-buffered async global->LDS staging (ASYNCcnt).
// ---------------------------------------------------------------------------

#define BATCH  64
#define NTOK   196
#define DIN    384
#define DOUT   1024
#define DSEC   512
#define SECIN  73920      // 384*385/2
#define SKETCH 2048
#define EFFSK  768
#define EPSV   1e-5f

typedef float v2f __attribute__((ext_vector_type(2)));
typedef float v8f __attribute__((ext_vector_type(8)));
typedef int   v4i __attribute__((ext_vector_type(4)));

// Workspace layout (float offsets). Peak ~153 MB.
#define OFF_TC   0                                 // 64*196*384 tc
#define OFF_WT   (OFF_TC + BATCH*NTOK*DIN)         // 64*196*384 wt
#define OFF_Z    (OFF_WT + BATCH*NTOK*DIN)         // 64*384*384 M2 then Z
#define OFF_YA   (OFF_Z  + BATCH*DIN*DIN)          // 64*384*384
#define OFF_YB   (OFF_YA + BATCH*DIN*DIN)          // 64*384*384
#define OFF_P    OFF_TC                            // reuse tc+wt region
#define OFF_VEC  OFF_TC                            // reuse again after NS
#define OFF_DINV (OFF_YB + BATCH*DIN*DIN)          // 64*196
#define OFF_TRW  (OFF_DINV + BATCH*NTOK)           // 64   (trw+EPS)
#define OFF_MU   (OFF_TRW + BATCH)                 // 64*384
#define OFF_WC   (OFF_MU + BATCH*DIN)              // 64*384
#define OFF_TRNS (OFF_WC + BATCH*DIN)              // 64   (tr(M2)+EPS)
#define OFF_SK   (OFF_TRNS + BATCH)                // 64*2048

__device__ __forceinline__ v8f wmma_f32(v2f a, v2f b, v8f c) {
  return __builtin_amdgcn_wmma_f32_16x16x4_f32(
      /*neg_a=*/false, a, /*neg_b=*/false, b,
      /*c_mod=*/(short)0, c, /*reuse_a=*/false, /*reuse_b=*/false);
}

// ---- async global->LDS copy (CDNA5 GLOBAL_LOAD_ASYNC_TO_LDS_B128) ---------
// Builtin signature (from clang diagnostic): (v4i addrspace(1)*, v4i
// addrspace(3)*, int offset, int cpol). Addrspace casts via integer
// round-trip; low 32 bits of a generic shared address == LDS offset.
#define AS1 __attribute__((address_space(1)))
#define AS3 __attribute__((address_space(3)))
#if __has_builtin(__builtin_amdgcn_global_load_async_to_lds_b128)
#define ASYNC_COPY_B128(g, l)                                                  \
  __builtin_amdgcn_global_load_async_to_lds_b128(                              \
      (AS1 v4i*)(unsigned long long)(const void*)(g),                          \
      (AS3 v4i*)(unsigned)(unsigned long long)(const void*)(l), 0, 0)
#else
#define ASYNC_COPY_B128(g, l)                                                  \
  asm volatile("global_load_async_to_lds_b128 %0, %1, off" ::                  \
               "v"((unsigned)(unsigned long long)(const void*)(l)),            \
               "v"((const void*)(g)) : "memory")
#endif

#if __has_builtin(__builtin_amdgcn_s_wait_asynccnt)
#define WAIT_ASYNC(n) __builtin_amdgcn_s_wait_asynccnt(n)
#else
#define WAIT_ASYNC(n) asm volatile("s_wait_asynccnt %0" :: "i"(n) : "memory")
#endif

// ---------------------------------------------------------------------------
// Kernel 1: per-batch stats. deg, dinv, row, trw, mu, tc, wc (closed form).
// ---------------------------------------------------------------------------
__global__ __launch_bounds__(256)
void stats_kernel(const float* __restrict__ tokens,
                  const float* __restrict__ graph,
                  float* __restrict__ ws) {
  const int b = blockIdx.x, tid = threadIdx.x;
  __shared__ float sdinv[NTOK];
  __shared__ float srow[NTOK];
  __shared__ float smu[DIN];
  __shared__ float sred[256];
  const float* g = graph + (size_t)b * NTOK * NTOK;

  if (tid < NTOK) {
    float deg = 0.f;
    for (int m = 0; m < NTOK; ++m) deg += g[tid * NTOK + m];
    float dv = rsqrtf(fmaxf(deg, EPSV));
    sdinv[tid] = dv;
    ws[OFF_DINV + b * NTOK + tid] = dv;
  }
  __syncthreads();
  if (tid < NTOK) {
    float s = 0.f;
    for (int m = 0; m < NTOK; ++m) s += g[tid * NTOK + m] * sdinv[m];
    srow[tid] = sdinv[tid] * s;
  }
  __syncthreads();
  sred[tid] = (tid < NTOK) ? g[tid * NTOK + tid] * sdinv[tid] * sdinv[tid] : 0.f;
  __syncthreads();
  for (int s = 128; s > 0; s >>= 1) { if (tid < s) sred[tid] += sred[tid + s]; __syncthreads(); }
  const float trwE = sred[0] + EPSV;
  __syncthreads();
  sred[tid] = (tid < NTOK) ? srow[tid] : 0.f;
  __syncthreads();
  for (int s = 128; s > 0; s >>= 1) { if (tid < s) sred[tid] += sred[tid + s]; __syncthreads(); }
  const float sumrow = sred[0];
  __syncthreads();
  if (tid == 0) ws[OFF_TRW + b] = trwE;

  const float* tk = tokens + (size_t)b * NTOK * DIN;
  for (int d = tid; d < DIN; d += 256) {
    float acc = 0.f;
    for (int n = 0; n < NTOK; ++n) acc += tk[n * DIN + d] * srow[n];
    float m = acc / trwE;
    smu[d] = m;
    ws[OFF_MU + b * DIN + d] = m;
    ws[OFF_WC + b * DIN + d] = m * (trwE - sumrow) / trwE;   // closed form
  }
  __syncthreads();
  float* tc = ws + OFF_TC + (size_t)b * NTOK * DIN;
  for (int idx = tid; idx < NTOK * DIN; idx += 256)
    tc[idx] = tk[idx] - smu[idx % DIN];
}

// ---------------------------------------------------------------------------
// Kernel 2: wt = W @ tc, W normalized on the fly. Wave-per-16x16-tile.
// ---------------------------------------------------------------------------
__global__ __launch_bounds__(32)
void wt_wmma_kernel(const float* __restrict__ graph, float* __restrict__ ws) {
  const int b = blockIdx.y;
  const int tileM = blockIdx.x % 13, tileN = blockIdx.x / 13;
  const int lane = threadIdx.x, half = lane >> 4, l16 = lane & 15;
  const float* g    = graph + (size_t)b * NTOK * NTOK;
  const float* dinv = ws + OFF_DINV + b * NTOK;
  const float* tc   = ws + OFF_TC + (size_t)b * NTOK * DIN;
  float*       wt   = ws + OFF_WT + (size_t)b * NTOK * DIN;

  const int mRow = tileM * 16 + l16;
  const int mC   = (mRow < NTOK) ? mRow : (NTOK - 1);
  const float dm = dinv[mC];
  const int nCol = tileN * 16 + l16;

  v8f acc = {};
  for (int ks = 0; ks < 49; ++ks) {
    const int kb = ks * 4 + half * 2;
    v2f a, bb;
    a.x  = g[mC * NTOK + kb]     * dm * dinv[kb];
    a.y  = g[mC * NTOK + kb + 1] * dm * dinv[kb + 1];
    bb.x = tc[kb * DIN + nCol];
    bb.y = tc[(kb + 1) * DIN + nCol];
    acc = wmma_f32(a, bb, acc);
  }
  for (int v = 0; v < 8; ++v) {
    const int m = tileM * 16 + v + half * 8;
    if (m < NTOK) wt[m * DIN + nCol] = acc[v];
  }
}

// ---------------------------------------------------------------------------
// Kernel 3: M2 = tc^T @ wt -> ws[Z]
// ---------------------------------------------------------------------------
__global__ __launch_bounds__(32)
void m2_wmma_kernel(float* __restrict__ ws) {
  const int b = blockIdx.y;
  const int tileM = blockIdx.x % 24, tileN = blockIdx.x / 24;
  const int lane = threadIdx.x, half = lane >> 4, l16 = lane & 15;
  const float* tc = ws + OFF_TC + (size_t)b * NTOK * DIN;
  const float* wt = ws + OFF_WT + (size_t)b * NTOK * DIN;
  float*       m2 = ws + OFF_Z  + (size_t)b * DIN * DIN;

  const int mRow = tileM * 16 + l16;
  const int nCol = tileN * 16 + l16;
  v8f acc = {};
  for (int ks = 0; ks < 49; ++ks) {
    const int kb = ks * 4 + half * 2;
    v2f a, bb;
    a.x  = tc[kb * DIN + mRow];
    a.y  = tc[(kb + 1) * DIN + mRow];
    bb.x = wt[kb * DIN + nCol];
    bb.y = wt[(kb + 1) * DIN + nCol];
    acc = wmma_f32(a, bb, acc);
  }
  for (int v = 0; v < 8; ++v)
    m2[(tileM * 16 + v + half * 8) * DIN + nCol] = acc[v];
}

// ---------------------------------------------------------------------------
// Kernel 4: NS prep. Z = M2/(tr+EPS) in place; YA = 1.5 I - 0.5 Z (iter 1).
// ---------------------------------------------------------------------------
__global__ __launch_bounds__(256)
void ns_prep_kernel(float* __restrict__ ws) {
  const int b = blockIdx.x, tid = threadIdx.x;
  __shared__ float sred[256];
  float* m2 = ws + OFF_Z + (size_t)b * DIN * DIN;
  float p = 0.f;
  for (int d = tid; d < DIN; d += 256) p += m2[d * DIN + d];
  sred[tid] = p; __syncthreads();
  for (int s = 128; s > 0; s >>= 1) { if (tid < s) sred[tid] += sred[tid + s]; __syncthreads(); }
  const float trE = sred[0] + EPSV;
  if (tid == 0) ws[OFF_TRNS + b] = trE;
  const float inv = 1.f / trE;
  float* ya = ws + OFF_YA + (size_t)b * DIN * DIN;
  for (int idx = tid; idx < DIN * DIN; idx += 256) {
    const float z = m2[idx] * inv;
    m2[idx] = z;
    const int i = idx / DIN, j = idx - i * DIN;
    ya[idx] = ((i == j) ? 1.5f : 0.f) - 0.5f * z;
  }
}

// ---------------------------------------------------------------------------
// Kernel 5: batched 384x384 fp32 WMMA GEMM, LDS-staged (+2 pad).
// C = alpha*(A@B) + beta*E
// ---------------------------------------------------------------------------
__global__ __launch_bounds__(256)
void ns_gemm_kernel(const float* __restrict__ Abase, const float* __restrict__ Bbase,
                    const float* __restrict__ Ebase, float* __restrict__ Cbase,
                    float alpha, float beta) {
  const int b = blockIdx.y;
  const int tm = blockIdx.x % 6, tn = blockIdx.x / 6;
  const int tid = threadIdx.x, w = tid >> 5, lane = tid & 31;
  const int half = lane >> 4, l16 = lane & 15;
  __shared__ float ldsA[64 * 66];
  __shared__ float ldsB[64 * 66];
  const size_t boff = (size_t)b * DIN * DIN;
  const float* A  = Abase + boff;
  const float* Bm = Bbase + boff;

  const int m0l = (w & 3) * 16;
  const int n0l = (w >> 2) * 32;
  v8f acc0 = {}, acc1 = {};

  for (int k0 = 0; k0 < DIN; k0 += 64) {
    for (int i = 0; i < 16; ++i) {
      const int idx = i * 256 + tid;
      const int r = idx >> 6, c = idx & 63;
      ldsA[r * 66 + c] = A[(tm * 64 + r) * DIN + k0 + c];
      ldsB[r * 66 + c] = Bm[(k0 + r) * DIN + tn * 64 + c];
    }
    __syncthreads();
    for (int ks = 0; ks < 16; ++ks) {
      const int kb = ks * 4 + half * 2;
      v2f a, b0, b1;
      a.x  = ldsA[(m0l + l16) * 66 + kb];
      a.y  = ldsA[(m0l + l16) * 66 + kb + 1];
      b0.x = ldsB[kb * 66 + n0l + l16];
      b0.y = ldsB[(kb + 1) * 66 + n0l + l16];
      b1.x = ldsB[kb * 66 + n0l + 16 + l16];
      b1.y = ldsB[(kb + 1) * 66 + n0l + 16 + l16];
      acc0 = wmma_f32(a, b0, acc0);
      acc1 = wmma_f32(a, b1, acc1);
    }
    __syncthreads();
  }

  float* C = Cbase + boff;
  for (int v = 0; v < 8; ++v) {
    const int m   = tm * 64 + m0l + v + half * 8;
    const int n0g = tn * 64 + n0l + l16;
    float c0 = alpha * acc0[v];
    float c1 = alpha * acc1[v];
    if (Ebase) {
      const float* E = Ebase + boff;
      c0 += beta * E[m * DIN + n0g];
      c1 += beta * E[m * DIN + n0g + 16];
    }
    C[m * DIN + n0g]      = c0;
    C[m * DIN + n0g + 16] = c1;
  }
}

// ---------------------------------------------------------------------------
// Kernel 6: extract upper triangle of M2n = Y/sqrt(tr+EPS) -> vec
// ---------------------------------------------------------------------------
__global__ __launch_bounds__(256)
void extract_vec_kernel(float* __restrict__ ws) {
  const long long t = (long long)blockIdx.x * blockDim.x + threadIdx.x;
  if (t >= (long long)BATCH * DIN * DIN) return;
  const int b = (int)(t / (DIN * DIN));
  const int rem = (int)(t - (long long)b * DIN * DIN);
  const int i = rem / DIN, j = rem - i * DIN;
  if (i > j) return;
  const float scale = rsqrtf(ws[OFF_TRNS + b]);
  const int vi = i * DIN - (i * (i - 1)) / 2 + (j - i);
  ws[OFF_VEC + (size_t)b * SECIN + vi] =
      ws[OFF_YA + (size_t)b * DIN * DIN + rem] * scale;
}

// ---------------------------------------------------------------------------
// Kernel 7: count-sketch via LDS float atomics (ds_add_f32), pad to 2048.
// ---------------------------------------------------------------------------
__global__ __launch_bounds__(256)
void sketch_kernel(const float* __restrict__ signs, const int* __restrict__ hash_idx,
                   float* __restrict__ ws) {
  const int b = blockIdx.x, tid = threadIdx.x;
  __shared__ float s[3 * EFFSK];
  for (int i = tid; i < 3 * EFFSK; i += 256) s[i] = 0.f;
  __syncthreads();
  const float* wc = ws + OFF_WC + b * DIN;
  for (int i = 0; i < 3; ++i)
    for (int t = tid; t < DIN; t += 256)
      atomicAdd(&s[i * EFFSK + hash_idx[i * DIN + t]], wc[t] * signs[i * DIN + t]);
  __syncthreads();
  float* sk = ws + OFF_SK + (size_t)b * SKETCH;
  for (int e = tid; e < SKETCH; e += 256)
    sk[e] = (e < EFFSK) ? s[e] * s[EFFSK + e] * s[2 * EFFSK + e] : 0.f;
}

// ---------------------------------------------------------------------------
// Kernel 8: head GEMM  out[:, off:off+512] = gelu(bn(A(64xK) @ W(512xK)^T+b))
// Double-buffered ASYNC global->LDS staging: the async engine streams chunk
// k+1 while WMMAs consume chunk k. Rows padded to 68 floats (272B = 17*16)
// so every B128 LDS write is 16B-aligned and fragment gathers are
// bank-conflict free (stride 68 -> bank step 4 across 16 lanes).
// Per lane per chunk: 4 A-segments + 2 B-segments = 6 async ops; async loads
// complete in order, so s_wait_asynccnt<=6 retires exactly the current chunk.
// ---------------------------------------------------------------------------
#define HPAD 68
__global__ __launch_bounds__(256)
void head_gemm_kernel(const float* __restrict__ Av, const float* __restrict__ Wm,
                      const float* __restrict__ bias, const float* __restrict__ gamma,
                      const float* __restrict__ betav, float* __restrict__ out,
                      int K, int outOff) {
  const int tid = threadIdx.x, w = tid >> 5, lane = tid & 31;
  const int half = lane >> 4, l16 = lane & 15;
  const int nBlock = blockIdx.x * 32;
  __shared__ __align__(16) float ldsA[2 * 64 * HPAD];  // 64 batch rows x 64 k
  __shared__ __align__(16) float ldsB[2 * 32 * HPAD];  // 32 W rows    x 64 k

  const int m0  = (w & 3) * 16;
  const int n0l = (w >> 2) * 16;
  v8f acc = {};
  const int nchunks = K / 64;

  // prologue: stage chunk 0 into buffer 0
  {
    const int k0 = 0;
#pragma unroll
    for (int i = 0; i < 4; ++i) {                      // A: 1024 16B segments
      const int s = i * 256 + tid;
      const int r = s >> 4, c4 = (s & 15) << 2;
      ASYNC_COPY_B128(Av + (size_t)r * K + k0 + c4, ldsA + r * HPAD + c4);
    }
#pragma unroll
    for (int i = 0; i < 2; ++i) {                      // B: 512 16B segments
      const int s = i * 256 + tid;
      const int r = s >> 4, c4 = (s & 15) << 2;
      ASYNC_COPY_B128(Wm + (size_t)(nBlock + r) * K + k0 + c4, ldsB + r * HPAD + c4);
    }
  }

  for (int ci = 0; ci < nchunks; ++ci) {
    const int cur = ci & 1;
    if (ci + 1 < nchunks) {                            // stage next chunk
      const int k0 = (ci + 1) * 64;
      float* aBuf = ldsA + (cur ^ 1) * 64 * HPAD;
      float* bBuf = ldsB + (cur ^ 1) * 32 * HPAD;
#pragma unroll
      for (int i = 0; i < 4; ++i) {
        const int s = i * 256 + tid;
        const int r = s >> 4, c4 = (s & 15) << 2;
        ASYNC_COPY_B128(Av + (size_t)r * K + k0 + c4, aBuf + r * HPAD + c4);
      }
#pragma unroll
      for (int i = 0; i < 2; ++i) {
        const int s = i * 256 + tid;
        const int r = s >> 4, c4 = (s & 15) << 2;
        ASYNC_COPY_B128(Wm + (size_t)(nBlock + r) * K + k0 + c4, bBuf + r * HPAD + c4);
      }
      WAIT_ASYNC(6);                                   // current chunk retired
    } else {
      WAIT_ASYNC(0);
    }
    __syncthreads();                                   // all waves' stages visible

    const float* aBuf = ldsA + cur * 64 * HPAD;
    const float* bBuf = ldsB + cur * 32 * HPAD;
#pragma unroll
    for (int ks = 0; ks < 16; ++ks) {
      const int kb = ks * 4 + half * 2;
      v2f a, bb;
      a.x  = aBuf[(m0 + l16) * HPAD + kb];
      a.y  = aBuf[(m0 + l16) * HPAD + kb + 1];
      bb.x = bBuf[(n0l + l16) * HPAD + kb];            // B[k][n] = W[n][k]
      bb.y = bBuf[(n0l + l16) * HPAD + kb + 1];
      acc = wmma_f32(a, bb, acc);
    }
    __syncthreads();                                   // done reading cur buffer
  }

  const float invs = rsqrtf(1.0f + EPSV);
  for (int v = 0; v < 8; ++v) {
    const int m = m0 + v + half * 8;
    const int n = nBlock + n0l + l16;
    float y = acc[v] + bias[n];
    y = y * invs * gamma[n] + betav[n];
    const float ge = 0.5f * y * (1.0f + erff(y * 0.70710678118654752f));
    out[(size_t)m * DOUT + outOff + n] = ge;
  }
}

// ---------------------------------------------------------------------------
extern "C" void kernel_launch(void* const* d_in, const int* in_sizes, int n_in,
                              void* d_out, int out_size, void* d_ws, size_t ws_size,
                              hipStream_t stream) {
  const float* tokens   = (const float*)d_in[0];
  const float* graph    = (const float*)d_in[1];
  const float* signs    = (const float*)d_in[2];
  const float* W2       = (const float*)d_in[3];
  const float* b2       = (const float*)d_in[4];
  const float* g2       = (const float*)d_in[5];
  const float* be2      = (const float*)d_in[6];
  const float* W3       = (const float*)d_in[7];
  const float* b3       = (const float*)d_in[8];
  const float* g3       = (const float*)d_in[9];
  const float* be3      = (const float*)d_in[10];
  const int*   hash_idx = (const int*)d_in[11];
  float* out = (float*)d_out;
  float* ws  = (float*)d_ws;

  stats_kernel<<<BATCH, 256, 0, stream>>>(tokens, graph, ws);
  wt_wmma_kernel<<<dim3(13 * 24, BATCH), 32, 0, stream>>>(graph, ws);
  m2_wmma_kernel<<<dim3(24 * 24, BATCH), 32, 0, stream>>>(ws);
  ns_prep_kernel<<<BATCH, 256, 0, stream>>>(ws);

  const float* Z  = ws + OFF_Z;
  float* YA = ws + OFF_YA;
  float* YB = ws + OFF_YB;
  float* P  = ws + OFF_P;
  ns_gemm_kernel<<<dim3(36, BATCH), 256, 0, stream>>>(Z,  YA, nullptr, P,  1.0f,  0.0f);
  ns_gemm_kernel<<<dim3(36, BATCH), 256, 0, stream>>>(YA, P,  YA,      YB, -0.5f, 1.5f);
  ns_gemm_kernel<<<dim3(36, BATCH), 256, 0, stream>>>(Z,  YB, nullptr, P,  1.0f,  0.0f);
  ns_gemm_kernel<<<dim3(36, BATCH), 256, 0, stream>>>(YB, P,  YB,      YA, -0.5f, 1.5f);

  extract_vec_kernel<<<(BATCH * DIN * DIN) / 256, 256, 0, stream>>>(ws);
  sketch_kernel<<<BATCH, 256, 0, stream>>>(signs, hash_idx, ws);

  head_gemm_kernel<<<DSEC / 32, 256, 0, stream>>>(ws + OFF_VEC, W2, b2, g2, be2,
                                                  out, SECIN, 0);
  head_gemm_kernel<<<DSEC / 32, 256, 0, stream>>>(ws + OFF_SK, W3, b3, g3, be3,
                                                  out, SKETCH, DSEC);
}